// Decoder_77644418777893
// MI455X (gfx1250) — compile-verified
//
#include <hip/hip_runtime.h>
#include <math.h>

// MI455X / gfx1250, wave32. Fully fused attention-decoder step.
// One block per batch row b; 256 threads = 8 wave32s.
//
// Phases (per block, barriers between):
//  0   mean_n emb[b]                      (HBM stream, 512KB -> lands in L2)
//  0.5 q = mean@W_fixed + ctx@W_step ;  t[h] = Wk_h @ q_h  (tiny matvecs)
//  1   compat[n,h] = emb[b,n]·t[h]      via V_WMMA_F32_16X16X4_F32 (L2 re-read)
//  2   masked softmax stats m[h], z[h]; p = exp(compat-m) in LDS
//  3   s[h] = sum_n p[n,h]·emb[b,n]      (L2 re-read, VALU FMA rank-update)
//  3.5 glimpse_in -> glimpse -> g2 = Wl@glimpse (tiny matvecs)
//  4   logits[n] = 10*tanh(emb[b,n]·g2 / sqrt(D)), mask -> -inf (L2 re-read)

typedef __attribute__((ext_vector_type(2))) float v2f;
typedef __attribute__((ext_vector_type(8))) float v8f;

#define NN 1000
#define DD 128
#define NPAD 1008          // 63 tiles of 16 rows
#define NEG_BIG (-3.0e38f)

__global__ __launch_bounds__(256) void decoder_fused_kernel(
    const float* __restrict__ emb,   // [B,N,D]
    const float* __restrict__ ctx,   // [B,2D]
    const float* __restrict__ Wn,    // [D,3D]  (Wk | Wv | Wl)
    const float* __restrict__ Wf,    // [D,D]
    const float* __restrict__ Ws,    // [2D,D]
    const float* __restrict__ Wo,    // [D,D]
    const unsigned char* __restrict__ mask, // [B,N] bool, 1 = feasible
    float* __restrict__ out)         // [B,N]
{
    __shared__ float compat[NPAD * 8];   // 32256 B : compat, then p=exp(..)
    __shared__ float tld[16 * 130];      // t^T padded to 16 "heads", bank-padded
    __shared__ float sld[8 * 128];       // attention-weighted emb sums
    __shared__ float smean[128], qld[128], gild[128], glld[128], g2ld[128];
    __shared__ float red[256];           // generic reduction scratch
    __shared__ float mh[8], zi[8];

    const int tid = threadIdx.x;
    const int b   = blockIdx.x;
    const float* embB = emb + (size_t)b * (NN * DD);
    const unsigned char* mk = mask + (size_t)b * NN;

    // ---------------- Phase 0: column sums -> mean ----------------
    {
        const int c = tid & 127, p0 = tid >> 7;
        float s = 0.f;
        for (int n = p0; n < NN; n += 2) {
            if ((n & 31) == p0)
                __builtin_prefetch(&embB[(size_t)(n + 64) * DD + c], 0, 1);
            s += embB[(size_t)n * DD + c];
        }
        red[tid] = s;
    }
    __syncthreads();
    if (tid < 128) smean[tid] = (red[tid] + red[tid + 128]) * (1.0f / NN);
    __syncthreads();

    // ---------------- Phase 0.5: q and t ----------------
    {
        const int c = tid & 127;
        float a = 0.f;
        if (tid < 128) {
            for (int j = 0; j < 128; ++j) a += smean[j] * Wf[j * 128 + c];
        } else {
            const float* cb = ctx + (size_t)b * 256;
            for (int j = 0; j < 256; ++j) a += cb[j] * Ws[j * 128 + c];
        }
        red[tid] = a;
    }
    __syncthreads();
    if (tid < 128) qld[tid] = red[tid] + red[tid + 128];
    __syncthreads();
    if (tid < 128) {
        const int r = tid;
        const float* wr = Wn + (size_t)r * 384;        // Wk row r
        for (int h = 0; h < 8; ++h) {
            float a = 0.f;
            #pragma unroll
            for (int jj = 0; jj < 16; ++jj) a += qld[h * 16 + jj] * wr[h * 16 + jj];
            tld[h * 130 + r] = a * 0.25f;              // fold 1/sqrt(dk)
        }
    } else {
        const int r = tid - 128;
        for (int h = 8; h < 16; ++h) tld[h * 130 + r] = 0.f;  // head padding
    }
    __syncthreads();

    // ---------------- Phase 1: compat via f32 WMMA ----------------
    {
        const int lane = tid & 31, wv = tid >> 5;
        const int lr   = lane & 15;
        const int koff = (lane >> 4) << 1;   // half-wave K offset: 0 or 2
        // B operand (t^T, K x 16 heads) is tile-invariant: hoist all 32 frags.
        v2f bfr[32];
        {
            const float* tp = &tld[lr * 130 + koff];
            #pragma unroll
            for (int i = 0; i < 32; ++i) { bfr[i].x = tp[4 * i]; bfr[i].y = tp[4 * i + 1]; }
        }
        const int rbase = (lane >> 4) << 3;  // D rows 0..7 | 8..15 per half-wave
        for (int tile = wv; tile < 63; tile += 8) {
            const int n0 = tile * 16;
            int nr = n0 + lr; if (nr > NN - 1) nr = NN - 1;   // clamp pad rows
            const v2f* rp = (const v2f*)(embB + (size_t)nr * DD + koff);
            v8f c8 = {};
            #pragma unroll
            for (int i = 0; i < 32; ++i) {
                v2f a = rp[2 * i];
                c8 = __builtin_amdgcn_wmma_f32_16x16x4_f32(
                        false, a, false, bfr[i], (short)0, c8, false, false);
            }
            // D layout: lane(<16)=head col, VGPRs 0..7 = rows n0+rbase+0..7
            if (lr < 8) {
                #pragma unroll
                for (int v = 0; v < 8; ++v) {
                    const int n = n0 + rbase + v;
                    float val = NEG_BIG;
                    if (n < NN && mk[n]) val = c8[v];
                    compat[n * 8 + lr] = val;
                }
            }
        }
    }
    __syncthreads();

    // ---------------- Phase 2: softmax stats ----------------
    for (int i = tid; i < 8 * 128; i += 256) sld[i] = 0.f;
    {
        const int h = tid & 7, seg = tid >> 3;
        float lm = NEG_BIG;
        for (int n = seg; n < NPAD; n += 32) lm = fmaxf(lm, compat[n * 8 + h]);
        red[seg * 8 + h] = lm;
    }
    __syncthreads();
    if (tid < 8) {
        float m = NEG_BIG;
        for (int s = 0; s < 32; ++s) m = fmaxf(m, red[s * 8 + tid]);
        mh[tid] = m;
    }
    __syncthreads();
    {
        const int h = tid & 7, seg = tid >> 3;
        const float m = mh[h];
        float z = 0.f;
        for (int n = seg; n < NPAD; n += 32) {
            const float e = __expf(compat[n * 8 + h] - m);
            compat[n * 8 + h] = e;            // compat now holds unnormalized p
            z += e;
        }
        red[seg * 8 + h] = z;
    }
    __syncthreads();
    if (tid < 8) {
        float z = 0.f;
        for (int s = 0; s < 32; ++s) z += red[s * 8 + tid];
        zi[tid] = 1.0f / z;
    }
    __syncthreads();

    // ---------------- Phase 3: s[h] = sum_n p[n,h] * emb[b,n] ----------------
    {
        const int lane = tid & 31, wv = tid >> 5;
        float acc[8][4];
        #pragma unroll
        for (int h = 0; h < 8; ++h)
            #pragma unroll
            for (int j = 0; j < 4; ++j) acc[h][j] = 0.f;
        for (int tile = wv; tile < 63; tile += 8) {
            const int n0 = tile * 16;
            const int ne = (n0 + 16 < NN) ? n0 + 16 : NN;
            for (int n = n0; n < ne; ++n) {
                const float4 ev = ((const float4*)(embB + (size_t)n * DD))[lane];
                const float* pn = &compat[n * 8];
                #pragma unroll
                for (int h = 0; h < 8; ++h) {
                    const float pv = pn[h];     // LDS broadcast
                    acc[h][0] += pv * ev.x; acc[h][1] += pv * ev.y;
                    acc[h][2] += pv * ev.z; acc[h][3] += pv * ev.w;
                }
            }
        }
        #pragma unroll
        for (int h = 0; h < 8; ++h)
            #pragma unroll
            for (int j = 0; j < 4; ++j)
                atomicAdd(&sld[h * 128 + 4 * lane + j], acc[h][j]);  // ds_add_f32
    }
    __syncthreads();

    // ---------------- Phase 3.5: glimpse_in -> glimpse -> g2 ----------------
    if (tid < 128) {                     // glimpse_in[j] = zinv * s[h(j)] . Wv[:,j]
        const int j = tid, h = j >> 4;
        const float* sh = &sld[h * 128];
        float a = 0.f;
        for (int r = 0; r < 128; ++r) a += sh[r] * Wn[(size_t)r * 384 + 128 + j];
        gild[j] = a * zi[h];
    }
    __syncthreads();
    if (tid < 128) {                     // glimpse = glimpse_in @ W_out
        const int c = tid;
        float a = 0.f;
        for (int j = 0; j < 128; ++j) a += gild[j] * Wo[j * 128 + c];
        glld[c] = a;
    }
    __syncthreads();
    if (tid < 128) {                     // g2 = (Wl @ glimpse) / sqrt(D)
        const int r = tid;
        const float* wr = Wn + (size_t)r * 384 + 256;
        float a = 0.f;
        for (int c = 0; c < 128; ++c) a += wr[c] * glld[c];
        g2ld[r] = a * 0.08838834764831845f;   // 1/sqrt(128)
    }
    __syncthreads();

    // ---------------- Phase 4: logits ----------------
    {
        float* outB = out + (size_t)b * NN;
        for (int n = tid; n < NN; n += 256) {
            float res;
            if (!mk[n]) {
                res = -INFINITY;
            } else {
                const float4* rp4 = (const float4*)(embB + (size_t)n * DD);
                float d = 0.f;
                #pragma unroll 8
                for (int i = 0; i < 32; ++i) {
                    const float4 e = rp4[i];
                    d += e.x * g2ld[4 * i]     + e.y * g2ld[4 * i + 1]
                       + e.z * g2ld[4 * i + 2] + e.w * g2ld[4 * i + 3];
                }
                res = 10.0f * tanhf(d);
            }
            outB[n] = res;
        }
    }
}

extern "C" void kernel_launch(void* const* d_in, const int* in_sizes, int n_in,
                              void* d_out, int out_size, void* d_ws, size_t ws_size,
                              hipStream_t stream) {
    (void)n_in; (void)out_size; (void)d_ws; (void)ws_size;
    const float* emb = (const float*)d_in[0];
    const float* ctx = (const float*)d_in[1];
    const float* Wn  = (const float*)d_in[2];
    const float* Wf  = (const float*)d_in[3];
    const float* Ws  = (const float*)d_in[4];
    const float* Wo  = (const float*)d_in[5];
    const unsigned char* mk = (const unsigned char*)d_in[6];
    float* out = (float*)d_out;

    const int B = in_sizes[1] / 256;     // context_feat is [B, 2*128]
    decoder_fused_kernel<<<B, 256, 0, stream>>>(emb, ctx, Wn, Wf, Ws, Wo, mk, out);
}